// PNANet_33930241638746
// MI455X (gfx1250) — compile-verified
//
#include <hip/hip_runtime.h>
#include <math.h>

#define NN 50000
#define NE 1600000
#define NG 64
#define DELTA_C 2.5749f
#define BN_EPS_C 1e-5f

typedef float v2f __attribute__((ext_vector_type(2)));
typedef float v8f __attribute__((ext_vector_type(8)));

// ---------------------------------------------------------------- utilities
__global__ void k_zero32(unsigned* __restrict__ p, int n) {
  for (int i = blockIdx.x * blockDim.x + threadIdx.x; i < n;
       i += gridDim.x * blockDim.x)
    p[i] = 0u;
}

__global__ void k_hist(const int* __restrict__ dst, int* __restrict__ cnt, int ne) {
  for (int e = blockIdx.x * blockDim.x + threadIdx.x; e < ne;
       e += gridDim.x * blockDim.x)
    atomicAdd(&cnt[dst[e]], 1);
}

// single-workgroup scan: rowptr[i] exclusive prefix, cursor[i]=rowptr[i]
__global__ void k_scan(const int* __restrict__ cnt, int* __restrict__ rowptr,
                       int* __restrict__ cursor, int n) {
  __shared__ int buf[1024];
  __shared__ int carry;
  const int tid = threadIdx.x;
  if (tid == 0) { carry = 0; rowptr[0] = 0; }
  __syncthreads();
  for (int base = 0; base < n; base += 1024) {
    int i = base + tid;
    int v = (i < n) ? cnt[i] : 0;
    buf[tid] = v;
    __syncthreads();
    for (int off = 1; off < 1024; off <<= 1) {
      int t = (tid >= off) ? buf[tid - off] : 0;
      __syncthreads();
      buf[tid] += t;
      __syncthreads();
    }
    int inc = buf[tid] + carry;
    if (i < n) { rowptr[i + 1] = inc; cursor[i] = inc - v; }
    int tot = buf[1023];
    __syncthreads();
    if (tid == 0) carry += tot;
    __syncthreads();
  }
}

__global__ void k_scatter(const int* __restrict__ src, const int* __restrict__ dst,
                          int* __restrict__ cursor, int* __restrict__ ssrc, int ne) {
  for (int e = blockIdx.x * blockDim.x + threadIdx.x; e < ne;
       e += gridDim.x * blockDim.x) {
    int p = atomicAdd(&cursor[dst[e]], 1);
    ssrc[p] = src[e];
  }
}

// Pair-packed transpose: WTP[(k>>1)*128 + 2n + (k&1)] = W[n*K+k]
// so the two adjacent-K weights a lane needs form one 8-byte load.
__global__ void k_transpose_pack(const float* __restrict__ W, float* __restrict__ WTP,
                                 int K) {
  int i = blockIdx.x * blockDim.x + threadIdx.x;
  if (i < K * 64) {
    int k = i >> 6, n = i & 63;
    WTP[(k >> 1) * 128 + (n << 1) + (k & 1)] = W[n * K + k];
  }
}

// --------------------------------------------------- per-node PNA aggregation
// one wave per node; lane handles feature dims {2*lane, 2*lane+1}
__global__ void k_aggr(const float* __restrict__ x, const int* __restrict__ rowptr,
                       const int* __restrict__ ssrc, float* __restrict__ aggr,
                       float* __restrict__ cntf, int nnodes) {
  int wave = (blockIdx.x * blockDim.x + threadIdx.x) >> 5;
  int lane = threadIdx.x & 31;
  if (wave >= nnodes) return;
  int beg = rowptr[wave], end = rowptr[wave + 1];
  v2f s = {0.f, 0.f}, q = {0.f, 0.f}, mx = {-3.4e38f, -3.4e38f};
  for (int e = beg; e < end; ++e) {
    int sidx = ssrc[e];
    v2f v = *(const v2f*)(x + (long)sidx * 64 + lane * 2);
    s.x += v.x; s.y += v.y;
    q.x += v.x * v.x; q.y += v.y * v.y;
    mx.x = fmaxf(mx.x, v.x); mx.y = fmaxf(mx.y, v.y);
  }
  float c = (float)(end - beg);
  float inv = 1.0f / c;
  v2f mean = {s.x * inv, s.y * inv};
  v2f var = {fmaxf(q.x * inv - mean.x * mean.x, 0.f),
             fmaxf(q.y * inv - mean.y * mean.y, 0.f)};
  float* arow = aggr + (long)wave * 256;
  *(v2f*)(arow + lane * 2)       = s;
  *(v2f*)(arow + 64 + lane * 2)  = mx;
  *(v2f*)(arow + 128 + lane * 2) = mean;
  *(v2f*)(arow + 192 + lane * 2) = var;
  if (lane == 0) cntf[wave] = c;
}

// --------------------------------------------------- MLP: a = comb @ W^T + b
// comb = [aggr, (c/D)*aggr, (D/c)*aggr]; computed as 3 K=256 WMMA chains.
// wtp = pair-packed transposed mlp weights. hout = x + a (16x64 tile / wave).
__global__ void __launch_bounds__(128)
k_mlp(const float* __restrict__ aggr, const float* __restrict__ cntf,
      const float* __restrict__ wtp, const float* __restrict__ bias,
      const float* __restrict__ xin, float* __restrict__ hout, int nnodes) {
  int wave = (blockIdx.x * blockDim.x + threadIdx.x) >> 5;
  int lane = threadIdx.x & 31;
  int rowBase = wave * 16;
  if (rowBase >= nnodes) return;
  int half = lane >> 4;   // 0 or 1
  int l16  = lane & 15;

  v8f acc[3][4];
#pragma unroll
  for (int s = 0; s < 3; ++s)
#pragma unroll
    for (int t = 0; t < 4; ++t)
      acc[s][t] = (v8f){0.f, 0.f, 0.f, 0.f, 0.f, 0.f, 0.f, 0.f};

  const float* arow = aggr + (long)(rowBase + l16) * 256 + (half << 1);
  for (int k = 0; k < 256; k += 4) {
    v2f a = *(const v2f*)(arow + k);
#pragma unroll
    for (int s = 0; s < 3; ++s) {
      // row-pair index: (k + s*256)/2 + half ; each frag = one b64 load
      const float* w0 = wtp + (long)((k >> 1) + (s << 7) + half) * 128;
#pragma unroll
      for (int t = 0; t < 4; ++t) {
        v2f b = *(const v2f*)(w0 + ((t * 16 + l16) << 1));
        acc[s][t] = __builtin_amdgcn_wmma_f32_16x16x4_f32(
            false, a, false, b, (short)0, acc[s][t], false, false);
      }
    }
  }
  // epilogue: combine PNA scalers per-row, add bias and residual x
  float cf[8];
#pragma unroll
  for (int v = 0; v < 8; ++v) cf[v] = cntf[rowBase + v + (half << 3)];
#pragma unroll
  for (int t = 0; t < 4; ++t) {
    int col = t * 16 + l16;
    float bc = bias[col];
#pragma unroll
    for (int v = 0; v < 8; ++v) {
      int grow = rowBase + v + (half << 3);
      float c = cf[v];
      float val = acc[0][t][v] + (c * (1.0f / DELTA_C)) * acc[1][t][v] +
                  (DELTA_C / c) * acc[2][t][v] + bc + xin[(long)grow * 64 + col];
      hout[(long)grow * 64 + col] = val;
    }
  }
}

// ------------------------------------- FFN: relu(lin2(relu(lin1(h)))) fused
__global__ void __launch_bounds__(128)
k_ffn(const float* __restrict__ h1, const float* __restrict__ w1p,
      const float* __restrict__ b1, const float* __restrict__ w2p,
      const float* __restrict__ b2, float* __restrict__ hout, int nnodes) {
  __shared__ float tile[4][16 * 64];   // one 16x64 tile per wave
  int wslot = threadIdx.x >> 5;
  int wave = (blockIdx.x * blockDim.x + threadIdx.x) >> 5;
  int lane = threadIdx.x & 31;
  int rowBase = wave * 16;
  if (rowBase >= nnodes) return;
  int half = lane >> 4, l16 = lane & 15;
  float* lt = tile[wslot];

  v8f acc[4];
#pragma unroll
  for (int t = 0; t < 4; ++t) acc[t] = (v8f){0.f,0.f,0.f,0.f,0.f,0.f,0.f,0.f};
  const float* arow = h1 + (long)(rowBase + l16) * 64 + (half << 1);
  for (int k = 0; k < 64; k += 4) {
    v2f a = *(const v2f*)(arow + k);
    const float* w0 = w1p + (long)((k >> 1) + half) * 128;
#pragma unroll
    for (int t = 0; t < 4; ++t) {
      v2f b = *(const v2f*)(w0 + ((t * 16 + l16) << 1));
      acc[t] = __builtin_amdgcn_wmma_f32_16x16x4_f32(
          false, a, false, b, (short)0, acc[t], false, false);
    }
  }
  // bias + relu -> per-wave LDS tile (C-layout -> row-major, A-layout source)
#pragma unroll
  for (int t = 0; t < 4; ++t) {
    int col = t * 16 + l16;
    float bb = b1[col];
#pragma unroll
    for (int v = 0; v < 8; ++v) {
      int m = v + (half << 3);
      float val = acc[t][v] + bb;
      lt[m * 64 + col] = val > 0.f ? val : 0.f;
    }
  }
  v8f acc2[4];
#pragma unroll
  for (int t = 0; t < 4; ++t) acc2[t] = (v8f){0.f,0.f,0.f,0.f,0.f,0.f,0.f,0.f};
  const float* arow2 = lt + l16 * 64 + (half << 1);
  for (int k = 0; k < 64; k += 4) {
    v2f a = *(const v2f*)(arow2 + k);
    const float* w0 = w2p + (long)((k >> 1) + half) * 128;
#pragma unroll
    for (int t = 0; t < 4; ++t) {
      v2f b = *(const v2f*)(w0 + ((t * 16 + l16) << 1));
      acc2[t] = __builtin_amdgcn_wmma_f32_16x16x4_f32(
          false, a, false, b, (short)0, acc2[t], false, false);
    }
  }
#pragma unroll
  for (int t = 0; t < 4; ++t) {
    int col = t * 16 + l16;
    float bb = b2[col];
#pragma unroll
    for (int v = 0; v < 8; ++v) {
      int m = v + (half << 3);
      float val = acc2[t][v] + bb;
      val = val > 0.f ? val : 0.f;               // outer relu before BN
      hout[(long)(rowBase + m) * 64 + col] = val;
    }
  }
}

// ----------------------------------------------------------- batchnorm stats
__global__ void k_colstats(const float* __restrict__ h, float* __restrict__ stats,
                           int nnodes) {
  int tid = blockIdx.x * blockDim.x + threadIdx.x;
  int col = tid & 63;
  int row0 = tid >> 6;
  int stride = (gridDim.x * blockDim.x) >> 6;
  float s = 0.f, q = 0.f;
  for (int r = row0; r < nnodes; r += stride) {
    float v = h[(long)r * 64 + col];
    s += v; q += v * v;
  }
  atomicAdd(&stats[col], s);
  atomicAdd(&stats[64 + col], q);
}

__global__ void k_bn(float* __restrict__ h, const float* __restrict__ stats,
                     const float* __restrict__ g, const float* __restrict__ b,
                     int nnodes) {
  const float invn = 1.0f / (float)NN;
  int total = nnodes * 64;
  for (int i = blockIdx.x * blockDim.x + threadIdx.x; i < total;
       i += gridDim.x * blockDim.x) {
    int col = i & 63;
    float mean = stats[col] * invn;
    float var = stats[64 + col] * invn - mean * mean;
    float inv = rsqrtf(var + BN_EPS_C);
    h[i] = g[col] * (h[i] - mean) * inv + b[col];
  }
}

// ------------------------------------------------------------- pooling, head
__global__ void k_pool(const float* __restrict__ h, const int* __restrict__ batch,
                       float* __restrict__ gout, int nnodes) {
  int total = nnodes * 64;
  for (int i = blockIdx.x * blockDim.x + threadIdx.x; i < total;
       i += gridDim.x * blockDim.x) {
    int node = i >> 6, col = i & 63;
    atomicAdd(&gout[batch[node] * 64 + col], h[i]);
  }
}

__global__ void k_head(const float* __restrict__ gpool, const float* __restrict__ fc1W,
                       const float* __restrict__ fc1b, const float* __restrict__ fc2W,
                       const float* __restrict__ fc2b, float* __restrict__ out) {
  __shared__ float G[64 * 64];
  __shared__ float H[64 * 64];
  __shared__ float O[64 * 32];
  int tid = threadIdx.x;  // 256 threads
  for (int i = tid; i < 4096; i += 256) G[i] = gpool[i];
  __syncthreads();
  for (int i = tid; i < 4096; i += 256) {
    int r = i >> 6, c = i & 63;
    float a = fc1b[c];
    for (int k = 0; k < 64; ++k) a += G[r * 64 + k] * fc1W[c * 64 + k];
    H[i] = a > 0.f ? a : 0.f;
  }
  __syncthreads();
  for (int i = tid; i < 2048; i += 256) {
    int r = i >> 5, c = i & 31;
    float a = fc2b[c];
    for (int k = 0; k < 64; ++k) a += H[r * 64 + k] * fc2W[c * 64 + k];
    O[i] = a;
  }
  __syncthreads();
  if (tid < 64) {
    int r = tid;
    float m = -3.4e38f;
    for (int c = 0; c < 32; ++c) m = fmaxf(m, O[r * 32 + c]);
    float sum = 0.f;
    for (int c = 0; c < 32; ++c) sum += expf(O[r * 32 + c] - m);
    float lse = m + logf(sum);
    for (int c = 0; c < 32; ++c) out[r * 32 + c] = O[r * 32 + c] - lse;
  }
}

// --------------------------------------------------------------------- host
extern "C" void kernel_launch(void* const* d_in, const int* in_sizes, int n_in,
                              void* d_out, int out_size, void* d_ws, size_t ws_size,
                              hipStream_t stream) {
  const float* x = (const float*)d_in[0];
  const int* ei = (const int*)d_in[1];
  const int* esrc = ei;
  const int* edst = ei + NE;
  const int* batch = (const int*)d_in[2];

  char* wsb = (char*)d_ws;
  size_t o = 0;
  auto take = [&](size_t bytes) -> void* {
    o = (o + 255) & ~(size_t)255;
    void* p = wsb + o;
    o += bytes;
    return p;
  };
  int* cnt_i   = (int*)take((size_t)NN * 4);
  int* rowptr  = (int*)take((size_t)(NN + 1) * 4);
  int* cursor  = (int*)take((size_t)NN * 4);
  int* ssrc    = (int*)take((size_t)NE * 4);
  float* cntf  = (float*)take((size_t)NN * 4);
  float* aggr  = (float*)take((size_t)NN * 256 * 4);
  float* bufT  = (float*)take((size_t)NN * 64 * 4);
  float* bufA  = (float*)take((size_t)NN * 64 * 4);
  float* bufB  = (float*)take((size_t)NN * 64 * 4);
  float* wtMLP = (float*)take((size_t)768 * 64 * 4);
  float* wt1   = (float*)take((size_t)64 * 64 * 4);
  float* wt2   = (float*)take((size_t)64 * 64 * 4);
  float* cstat = (float*)take(128 * 4);
  float* gpool = (float*)take((size_t)NG * 64 * 4);

  // ---- CSR build (edges constant across layers)
  k_zero32<<<256, 256, 0, stream>>>((unsigned*)cnt_i, NN);
  k_hist<<<6250, 256, 0, stream>>>(edst, cnt_i, NE);
  k_scan<<<1, 1024, 0, stream>>>(cnt_i, rowptr, cursor, NN);
  k_scatter<<<6250, 256, 0, stream>>>(esrc, edst, cursor, ssrc, NE);

  const float* cur = x;
  float* outs[3] = {bufA, bufB, bufA};
  for (int l = 0; l < 3; ++l) {
    const float* mlpW = (const float*)d_in[3 + 6 * l];
    const float* mlpb = (const float*)d_in[4 + 6 * l];
    const float* W1   = (const float*)d_in[5 + 6 * l];
    const float* b1   = (const float*)d_in[6 + 6 * l];
    const float* W2   = (const float*)d_in[7 + 6 * l];
    const float* b2   = (const float*)d_in[8 + 6 * l];
    const float* bng  = (const float*)d_in[21 + 2 * l];
    const float* bnb  = (const float*)d_in[22 + 2 * l];
    float* hout = outs[l];

    k_transpose_pack<<<192, 256, 0, stream>>>(mlpW, wtMLP, 768);
    k_transpose_pack<<<16, 256, 0, stream>>>(W1, wt1, 64);
    k_transpose_pack<<<16, 256, 0, stream>>>(W2, wt2, 64);

    k_aggr<<<6250, 256, 0, stream>>>(cur, rowptr, ssrc, aggr, cntf, NN);
    k_mlp<<<782, 128, 0, stream>>>(aggr, cntf, wtMLP, mlpb, cur, bufT, NN);
    k_ffn<<<782, 128, 0, stream>>>(bufT, wt1, b1, wt2, b2, hout, NN);

    k_zero32<<<1, 128, 0, stream>>>((unsigned*)cstat, 128);
    k_colstats<<<256, 256, 0, stream>>>(hout, cstat, NN);
    k_bn<<<2048, 256, 0, stream>>>(hout, cstat, bng, bnb, NN);
    cur = hout;
  }

  const float* fc1W = (const float*)d_in[27];
  const float* fc1b = (const float*)d_in[28];
  const float* fc2W = (const float*)d_in[29];
  const float* fc2b = (const float*)d_in[30];

  k_zero32<<<16, 256, 0, stream>>>((unsigned*)gpool, NG * 64);
  k_pool<<<2048, 256, 0, stream>>>(cur, batch, gpool, NN);
  k_head<<<1, 256, 0, stream>>>(gpool, fc1W, fc1b, fc2W, fc2b, (float*)d_out);
}